// S4Layer_60739427500048
// MI455X (gfx1250) — compile-verified
//
#include <hip/hip_runtime.h>

// ---------------------------------------------------------------------------
// S4 layer for MI455X (gfx1250).
// FFT convolution: every 4096-pt DFT is factored 64x64; the 64^3 complex
// matmuls run on V_WMMA_F32_16X16X4_F32. LDS holds split re/im planes (SoA)
// so WMMA fragments load directly as 2-dword LDS reads with no repacking.
// Inverse transform uses IDFT(Z) = conj(DFT(conj Z))/N, so only one (forward)
// DFT path exists. The hot conv kernel stages u -> LDS and LDS -> y with the
// Tensor Data Mover (tensor_load_to_lds / tensor_store_from_lds, TENSORcnt).
// ---------------------------------------------------------------------------

typedef float v2f  __attribute__((ext_vector_type(2)));
typedef float v8f  __attribute__((ext_vector_type(8)));
typedef unsigned int uint32x4 __attribute__((ext_vector_type(4)));
typedef int          int32x8  __attribute__((ext_vector_type(8)));
typedef int          int32x4  __attribute__((ext_vector_type(4)));

#define SEQ_L   2048
#define NFFT    4096
#define DMODEL  512
#define NBATCH  8
#define NPOLES  64
#define PI_F    3.14159265358979323846f

#define HAS_TDM (__has_builtin(__builtin_amdgcn_tensor_load_to_lds) && \
                 __has_builtin(__builtin_amdgcn_tensor_store_from_lds) && \
                 __has_builtin(__builtin_amdgcn_s_wait_tensorcnt))

#if HAS_TDM
// ---------------------------------------------------------------------------
// 1-D TDM copy: nelem contiguous 4-byte elements between global and LDS.
// D# layout per CDNA5 ISA ch.8 (group0: count/lds_addr/global_addr/type=2;
// group1: data_size=4B, tensor_dim0=tile_dim0=nelem, dim1=1, stride=nelem).
// Groups 2/3 unused (<=2D tensor). This toolchain's builtin takes 6 args
// (g0, g1, g2, g3, extra group, cpol) -- zero-fill the unused groups.
// Issue from ONE wave only (TDM ignores EXEC); wait with s_wait_tensorcnt.
// ---------------------------------------------------------------------------
__device__ __forceinline__ void tdm_copy_1d(unsigned lds_off, const void* gptr,
                                            unsigned nelem, bool to_lds) {
  unsigned long long ga = (unsigned long long)(uintptr_t)gptr;
  uint32x4 g0;
  g0[0] = 1u;                                            // count=1, user mode
  g0[1] = lds_off;                                       // lds_addr (bytes)
  g0[2] = (unsigned)(ga & 0xFFFFFFFFu);                  // global_addr[31:0]
  g0[3] = (unsigned)((ga >> 32) & 0x01FFFFFFu) | 0x80000000u; // [56:32]|type=2
  int32x8 g1;
  g1[0] = (int)(2u << 16);                               // data_size = 4 bytes
  g1[1] = (int)((nelem & 0xFFFFu) << 16);                // tensor_dim0[15:0]
  g1[2] = (int)((nelem >> 16) & 0xFFFFu) | (1 << 16);    // dim0[31:16], dim1=1
  g1[3] = (int)((nelem & 0xFFFFu) << 16);                // tile_dim0
  g1[4] = 1;                                             // tile_dim1 = 1
  g1[5] = (int)nelem;                                    // dim0_stride[31:0]
  g1[6] = 0;
  g1[7] = 0;
  int32x4 z4 = {0, 0, 0, 0};
  int32x8 z8 = {0, 0, 0, 0, 0, 0, 0, 0};
  if (to_lds)
    __builtin_amdgcn_tensor_load_to_lds(g0, g1, z4, z4, z8, 0);
  else
    __builtin_amdgcn_tensor_store_from_lds(g0, g1, z4, z4, z8, 0);
}
#endif

// ---------------------------------------------------------------------------
// Workgroup-cooperative 4096-point FORWARD DFT, radix-64 x 64, WMMA f32.
// SoA planes: xr/xi (in & out), yr/yi (ping), Fr/Fi = W64^{bc} table.
// Caller must __syncthreads() after filling x / F. Result back in xr/xi.
// ---------------------------------------------------------------------------
__device__ __forceinline__ void dft4096(float* __restrict__ xr, float* __restrict__ xi,
                                        float* __restrict__ yr, float* __restrict__ yi,
                                        const float* __restrict__ Fr,
                                        const float* __restrict__ Fi,
                                        float scale) {
  const int tid   = threadIdx.x;
  const int wave  = tid >> 5;
  const int lane  = tid & 31;
  const int row   = lane & 15;      // M-row for A/C, N-col for B/C
  const int khalf = lane >> 4;      // 0: K={0,1}, 1: K={2,3}

  // ---- Stage 1: Y[a,c] = sum_b x[a + 64b] * F[b,c], then twiddle W4096^{ac}
  for (int t = 0; t < 2; ++t) {
    const int tile = wave * 2 + t;
    const int at   = (tile >> 2) * 16;
    const int ct   = (tile & 3)  * 16;
    v8f rr = {0,0,0,0,0,0,0,0};
    v8f ii = {0,0,0,0,0,0,0,0};
    v8f ri = {0,0,0,0,0,0,0,0};
    v8f ir = {0,0,0,0,0,0,0,0};
#pragma unroll
    for (int kc = 0; kc < 16; ++kc) {
      const int k0 = kc * 4 + khalf * 2;
      const int ax = (at + row) + 64 * k0;        // A addresses: 256 B apart
      const int bx = k0 * 64 + (ct + row);        // B addresses: 256 B apart
      v2f Ar = { xr[ax], xr[ax + 64] };
      v2f Ai = { xi[ax], xi[ax + 64] };
      v2f Br = { Fr[bx], Fr[bx + 64] };
      v2f Bi = { Fi[bx], Fi[bx + 64] };
      rr = __builtin_amdgcn_wmma_f32_16x16x4_f32(false, Ar, false, Br, (short)0, rr, false, false);
      ii = __builtin_amdgcn_wmma_f32_16x16x4_f32(false, Ai, false, Bi, (short)0, ii, false, false);
      ri = __builtin_amdgcn_wmma_f32_16x16x4_f32(false, Ar, false, Bi, (short)0, ri, false, false);
      ir = __builtin_amdgcn_wmma_f32_16x16x4_f32(false, Ai, false, Br, (short)0, ir, false, false);
    }
#pragma unroll
    for (int i = 0; i < 8; ++i) {
      const int a = at + i + khalf * 8;
      const int c = ct + row;
      const float re = rr[i] - ii[i];
      const float im = ri[i] + ir[i];
      float sa, ca;
      __sincosf((-2.0f * PI_F / 4096.0f) * (float)(a * c), &sa, &ca);
      yr[a * 64 + c] = re * ca - im * sa;
      yi[a * 64 + c] = re * sa + im * ca;
    }
  }
  __syncthreads();

  // ---- Stage 3: X[c + 64*dd] = sum_a F[dd,a] * Z[a,c]
  for (int t = 0; t < 2; ++t) {
    const int tile = wave * 2 + t;
    const int dt   = (tile >> 2) * 16;
    const int ct   = (tile & 3)  * 16;
    v8f rr = {0,0,0,0,0,0,0,0};
    v8f ii = {0,0,0,0,0,0,0,0};
    v8f ri = {0,0,0,0,0,0,0,0};
    v8f ir = {0,0,0,0,0,0,0,0};
#pragma unroll
    for (int kc = 0; kc < 16; ++kc) {
      const int k0 = kc * 4 + khalf * 2;
      const int axi = (dt + row) * 64 + k0;       // A addresses: contiguous
      const int bx  = k0 * 64 + (ct + row);       // B addresses: 256 B apart
      v2f Ar = { Fr[axi], Fr[axi + 1] };
      v2f Ai = { Fi[axi], Fi[axi + 1] };
      v2f Br = { yr[bx], yr[bx + 64] };
      v2f Bi = { yi[bx], yi[bx + 64] };
      rr = __builtin_amdgcn_wmma_f32_16x16x4_f32(false, Ar, false, Br, (short)0, rr, false, false);
      ii = __builtin_amdgcn_wmma_f32_16x16x4_f32(false, Ai, false, Bi, (short)0, ii, false, false);
      ri = __builtin_amdgcn_wmma_f32_16x16x4_f32(false, Ar, false, Bi, (short)0, ri, false, false);
      ir = __builtin_amdgcn_wmma_f32_16x16x4_f32(false, Ai, false, Br, (short)0, ir, false, false);
    }
#pragma unroll
    for (int i = 0; i < 8; ++i) {
      const int dd = dt + i + khalf * 8;
      const int c  = ct + row;
      xr[c + 64 * dd] = (rr[i] - ii[i]) * scale;
      xi[c + 64 * dd] = (ri[i] + ir[i]) * scale;
    }
  }
  __syncthreads();
}

// F[b*64+c] = exp(-2*pi*i*b*c/64) split into Fr/Fi planes
__device__ __forceinline__ void build_F(float* Fr, float* Fi) {
  for (int i = threadIdx.x; i < 64 * 64; i += 256) {
    const int b = i >> 6, c = i & 63;
    float s, ca;
    __sincosf((-2.0f * PI_F / 64.0f) * (float)((b * c) & 63), &s, &ca);
    Fr[i] = ca;
    Fi[i] = s;
  }
}

// ---------------------------------------------------------------------------
// Kernel 1: Cauchy resolvent -> at_roots[d, j]  (complex, VALU-bound)
// ---------------------------------------------------------------------------
__global__ void cauchy_kernel(const float* __restrict__ B, const float* __restrict__ Ct,
                              const float* __restrict__ log_step,
                              const float2* __restrict__ p, const float2* __restrict__ q,
                              const float2* __restrict__ lmbda,
                              const float2* __restrict__ omega,
                              float2* __restrict__ at_roots) {
  const int j = blockIdx.x * blockDim.x + threadIdx.x;   // 0..2047
  const int d = blockIdx.y;                              // 0..511
  if (j >= SEQ_L) return;

  const float2 w = omega[j];
  const float step = __expf(log_step[d]);
  const float opx = 1.0f + w.x, opy = w.y;               // 1 + omega
  const float omx = 1.0f - w.x, omy = -w.y;              // 1 - omega
  const float inv = 1.0f / (opx * opx + opy * opy);
  const float gsc = 2.0f / step;
  const float gr = (omx * opx + omy * opy) * inv * gsc;  // g = (2/step)(1-w)/(1+w)
  const float gi = (omy * opx - omx * opy) * inv * gsc;

  float k00r = 0.f, k00i = 0.f, k01r = 0.f, k01i = 0.f;
  float k10r = 0.f, k10i = 0.f, k11r = 0.f, k11i = 0.f;
#pragma unroll 4
  for (int n = 0; n < NPOLES; ++n) {
    const float2 lam = lmbda[n];
    const float dr = gr - lam.x, di = gi - lam.y;
    const float rinv = 1.0f / (dr * dr + di * di);
    const float rr = dr * rinv, rim = -di * rinv;        // r = 1/(g - lambda)
    const float Bn = B[d * NPOLES + n];
    const float Cn = Ct[d * NPOLES + n];                 // conj(Ct)=Ct (real)
    const float2 pn = p[n];
    const float2 qn = q[n];
    const float a1x = qn.x, a1y = -qn.y;                 // conj(q)
    const float w00 = Cn * Bn;
    const float w01x = Cn * pn.x,  w01y = Cn * pn.y;
    const float w10x = a1x * Bn,   w10y = a1y * Bn;
    const float w11x = a1x * pn.x - a1y * pn.y;
    const float w11y = a1x * pn.y + a1y * pn.x;
    k00r += w00 * rr;                  k00i += w00 * rim;
    k01r += w01x * rr - w01y * rim;    k01i += w01x * rim + w01y * rr;
    k10r += w10x * rr - w10y * rim;    k10i += w10x * rim + w10y * rr;
    k11r += w11x * rr - w11y * rim;    k11i += w11x * rim + w11y * rr;
  }
  const float dnx = 1.0f + k11r, dny = k11i;
  const float dinv = 1.0f / (dnx * dnx + dny * dny);
  const float numx = k01r * k10r - k01i * k10i;
  const float numy = k01r * k10i + k01i * k10r;
  const float frx = (numx * dnx + numy * dny) * dinv;
  const float fry = (numy * dnx - numx * dny) * dinv;
  const float tx = k00r - frx, ty = k00i - fry;
  const float cr = 2.0f * opx * inv, ci = -2.0f * opy * inv;
  at_roots[(size_t)d * SEQ_L + j] = make_float2(tx * cr - ty * ci, tx * ci + ty * cr);
}

// ---------------------------------------------------------------------------
// Kernel 2: per-channel filter spectrum.
//   at_roots -> even slots -> DFT4096 -> K = Re/2048 (pad) -> DFT4096 -> Khat
// ---------------------------------------------------------------------------
__global__ void kernel_spectrum(const float2* __restrict__ at_roots,
                                float2* __restrict__ Khat) {
  extern __shared__ float smem[];
  float* xr = smem;
  float* xi = smem + NFFT;
  float* yr = smem + 2 * NFFT;
  float* yi = smem + 3 * NFFT;
  float* Fr = smem + 4 * NFFT;
  float* Fi = smem + 5 * NFFT;
  const int d = blockIdx.x;
  const int tid = threadIdx.x;

  build_F(Fr, Fi);
  for (int j = tid; j < SEQ_L; j += 256) {
    const float2 a = at_roots[(size_t)d * SEQ_L + j];
    xr[2 * j] = a.x;  xi[2 * j] = a.y;
    xr[2 * j + 1] = 0.f;  xi[2 * j + 1] = 0.f;
  }
  __syncthreads();

  dft4096(xr, xi, yr, yi, Fr, Fi, 1.0f);        // x = forward DFT (A_l)

  for (int l = tid; l < NFFT; l += 256) {       // K[l] = Re/2048, pad to 4096
    const float v = (l < SEQ_L) ? xr[l] * (1.0f / (float)SEQ_L) : 0.0f;
    xr[l] = v;
    xi[l] = 0.f;
  }
  __syncthreads();

  dft4096(xr, xi, yr, yi, Fr, Fi, 1.0f);        // Khat

  for (int f = tid; f < NFFT; f += 256)
    Khat[(size_t)d * NFFT + f] = make_float2(xr[f], xi[f]);
}

// ---------------------------------------------------------------------------
// Kernel 3: LDS-tiled transpose  u[b, l, d] -> ut[b, d, l]
// ---------------------------------------------------------------------------
__global__ void transpose_u(const float* __restrict__ u, float* __restrict__ ut) {
  __shared__ float tile[32][33];
  const int b  = blockIdx.z;
  const int l0 = blockIdx.x * 32;
  const int d0 = blockIdx.y * 32;
  const int tx = threadIdx.x & 31;
  const int ty = threadIdx.x >> 5;
  for (int i = ty; i < 32; i += 8)
    tile[i][tx] = u[((size_t)b * SEQ_L + (l0 + i)) * DMODEL + d0 + tx];
  __syncthreads();
  for (int i = ty; i < 32; i += 8)
    ut[((size_t)b * DMODEL + (d0 + i)) * SEQ_L + l0 + tx] = tile[tx][i];
}

// ---------------------------------------------------------------------------
// Kernel 4: FFT convolution of one (b,d) sequence, fully in LDS.
// TDM stages u into LDS and drains the result; inverse = conj/fwd/conj trick.
// ---------------------------------------------------------------------------
__global__ void conv_kernel(const float* __restrict__ ut,
                            const float2* __restrict__ Khat,
                            float* __restrict__ yt) {
  extern __shared__ float smem[];
  float* xr = smem;
  float* xi = smem + NFFT;
  float* yr = smem + 2 * NFFT;
  float* yi = smem + 3 * NFFT;
  float* Fr = smem + 4 * NFFT;
  float* Fi = smem + 5 * NFFT;
  const int seq = blockIdx.x;                    // b*512 + d
  const int d   = seq & (DMODEL - 1);
  const int tid = threadIdx.x;
  const int wave = tid >> 5;
  const float2* kh = Khat + (size_t)d * NFFT;
  const float*  up = ut + (size_t)seq * SEQ_L;
  float* yp = yt + (size_t)seq * SEQ_L;

  __builtin_prefetch(kh + tid * 16, 0, 1);       // warm L2 for the spectrum

#if HAS_TDM
  if (wave == 0)                                 // one DMA for the workgroup
    tdm_copy_1d((unsigned)(uintptr_t)xr, up, SEQ_L, true);
  for (int l = tid; l < NFFT; l += 256) {        // zero imag + zero-pad tail
    xi[l] = 0.f;
    if (l >= SEQ_L) xr[l] = 0.f;
  }
  build_F(Fr, Fi);
  if (wave == 0) __builtin_amdgcn_s_wait_tensorcnt(0);
  __syncthreads();
#else
  build_F(Fr, Fi);
  for (int l = tid; l < NFFT; l += 256) {
    xr[l] = (l < SEQ_L) ? up[l] : 0.0f;
    xi[l] = 0.f;
  }
  __syncthreads();
#endif

  dft4096(xr, xi, yr, yi, Fr, Fi, 1.0f);         // Uhat

  // pointwise Z = Uhat * Khat, stored conjugated: IDFT(Z)=conj(DFT(conj Z))/N
  for (int f = tid; f < NFFT; f += 256) {
    const float ax = xr[f], ay = xi[f];
    const float2 k = kh[f];
    xr[f] = ax * k.x - ay * k.y;
    xi[f] = -(ax * k.y + ay * k.x);
  }
  __syncthreads();

  dft4096(xr, xi, yr, yi, Fr, Fi, 1.0f / (float)NFFT);
  // y[l] = Re(conj(result)) = xr[l]; keep first 2048 (causal part)

#if HAS_TDM
  if (wave == 0) {
    tdm_copy_1d((unsigned)(uintptr_t)xr, yp, SEQ_L, false);
    __builtin_amdgcn_s_wait_tensorcnt(0);
  }
#else
  for (int l = tid; l < SEQ_L; l += 256)
    yp[l] = xr[l];
#endif
}

// ---------------------------------------------------------------------------
// Kernel 5: out[b, l, d] = yt[b, d, l] + D[d] * u[b, l, d]
// ---------------------------------------------------------------------------
__global__ void finalize_kernel(const float* __restrict__ yt,
                                const float* __restrict__ u,
                                const float* __restrict__ Dv,
                                float* __restrict__ out) {
  __shared__ float tile[32][33];
  const int b  = blockIdx.z;
  const int l0 = blockIdx.x * 32;
  const int d0 = blockIdx.y * 32;
  const int tx = threadIdx.x & 31;
  const int ty = threadIdx.x >> 5;
  for (int i = ty; i < 32; i += 8)
    tile[i][tx] = yt[((size_t)b * DMODEL + (d0 + i)) * SEQ_L + l0 + tx];
  __syncthreads();
  for (int i = ty; i < 32; i += 8) {
    const int l = l0 + i, dd = d0 + tx;
    const size_t idx = ((size_t)b * SEQ_L + l) * DMODEL + dd;
    out[idx] = tile[tx][i] + Dv[dd] * u[idx];
  }
}

// ---------------------------------------------------------------------------
// Launch: inputs in setup_inputs() order:
//   0:u 1:B 2:Ct 3:D 4:log_step 5:p 6:q 7:lmbda 8:omega_l
// ---------------------------------------------------------------------------
extern "C" void kernel_launch(void* const* d_in, const int* in_sizes, int n_in,
                              void* d_out, int out_size, void* d_ws, size_t ws_size,
                              hipStream_t stream) {
  const float*  u        = (const float*)d_in[0];
  const float*  B        = (const float*)d_in[1];
  const float*  Ct       = (const float*)d_in[2];
  const float*  Dv       = (const float*)d_in[3];
  const float*  log_step = (const float*)d_in[4];
  const float2* p        = (const float2*)d_in[5];
  const float2* q        = (const float2*)d_in[6];
  const float2* lmbda    = (const float2*)d_in[7];
  const float2* omega    = (const float2*)d_in[8];
  float* out = (float*)d_out;

  char* ws = (char*)d_ws;
  float2* at_roots = (float2*)(ws);                                   //  8 MB
  float2* Khat     = (float2*)(ws + (size_t)8  * 1024 * 1024);        // 16 MB
  float*  ut       = (float*) (ws + (size_t)24 * 1024 * 1024);        // 32 MB
  float*  yt       = (float*) (ws + (size_t)56 * 1024 * 1024);        // 32 MB

  const size_t smem = (size_t)6 * NFFT * sizeof(float);               // 96 KB LDS

  cauchy_kernel<<<dim3(SEQ_L / 256, DMODEL), 256, 0, stream>>>(
      B, Ct, log_step, p, q, lmbda, omega, at_roots);

  kernel_spectrum<<<DMODEL, 256, smem, stream>>>(at_roots, Khat);

  transpose_u<<<dim3(SEQ_L / 32, DMODEL / 32, NBATCH), 256, 0, stream>>>(u, ut);

  conv_kernel<<<NBATCH * DMODEL, 256, smem, stream>>>(ut, Khat, yt);

  finalize_kernel<<<dim3(SEQ_L / 32, DMODEL / 32, NBATCH), 256, 0, stream>>>(
      yt, u, Dv, out);
}